// RelationAwareBias_22943715295407
// MI455X (gfx1250) — compile-verified
//
#include <hip/hip_runtime.h>

typedef __attribute__((ext_vector_type(2))) float v2f;
typedef __attribute__((ext_vector_type(8))) float v8f;

#define HIDDEN 512
#define FEAT 4

// ---------------------------------------------------------------------------
// Kernel 1: fold the hidden dimension.
// wmean[f] = (1/512) * sum_h W[h][f],  bmean = (1/512) * sum_h b[h]
// Single wave32, deterministic shfl-tree reduction, writes 5 floats to ws.
// ---------------------------------------------------------------------------
__global__ void fold_hidden_kernel(const float* __restrict__ W,
                                   const float* __restrict__ bias,
                                   float* __restrict__ ws) {
    const int lane = threadIdx.x;  // 0..31
    float s0 = 0.f, s1 = 0.f, s2 = 0.f, s3 = 0.f, sb = 0.f;
    for (int r = lane; r < HIDDEN; r += 32) {
        const float4 w = *reinterpret_cast<const float4*>(W + r * FEAT);
        s0 += w.x; s1 += w.y; s2 += w.z; s3 += w.w;
        sb += bias[r];
    }
    #pragma unroll
    for (int off = 16; off > 0; off >>= 1) {
        s0 += __shfl_down(s0, off, 32);
        s1 += __shfl_down(s1, off, 32);
        s2 += __shfl_down(s2, off, 32);
        s3 += __shfl_down(s3, off, 32);
        sb += __shfl_down(sb, off, 32);
    }
    if (lane == 0) {
        const float inv = 1.0f / (float)HIDDEN;
        ws[0] = s0 * inv;
        ws[1] = s1 * inv;
        ws[2] = s2 * inv;
        ws[3] = s3 * inv;
        ws[4] = sb * inv;
    }
}

// ---------------------------------------------------------------------------
// Kernel 2: out[p] = dot(geom[p][0..3], wmean) + bmean via WMMA f32 16x16x4.
//
// Per wave: A (16x4) = geom vectors of 16 consecutive outputs
//   A layout (32-bit A 16x4): VGPR0 = {K0 | K2}, VGPR1 = {K1 | K3},
//   lanes 0-15 -> M, lanes 16-31 -> same M, high K half.
//   => lane loads geom[(base+m)*4 + kh*2] as b64; wave covers 256 contiguous B.
// B (4x16) = wmean broadcast into every column -> all D columns identical.
// D (16x16 f32): lane 0 holds M=0..7 (VGPR 0..7, N=0), lane 16 holds M=8..15.
//   Those two lanes store 2x float4 each (contiguous 64 B per wave).
// ---------------------------------------------------------------------------
__global__ void relation_bias_kernel(const float* __restrict__ geom,
                                     const float* __restrict__ ws,
                                     float* __restrict__ out) {
    const int tid  = threadIdx.x;
    const int lane = tid & 31;
    const int wave = tid >> 5;
    const int m    = lane & 15;   // output row within the 16-output tile
    const int kh   = lane >> 4;   // K half: 0 -> K{0,1}, 1 -> K{2,3}

    // 16 outputs per wave, 8 waves per block -> 128 outputs per block
    const long base = ((long)blockIdx.x * 8 + wave) * 16;

    // ---- A matrix: one b64 load per lane, fully coalesced ----
    const float2 av =
        *reinterpret_cast<const float2*>(geom + (base + m) * FEAT + kh * 2);
    v2f a; a.x = av.x; a.y = av.y;

    // ---- B matrix: wmean replicated across all 16 columns ----
    const float w0 = ws[0], w1 = ws[1], w2 = ws[2], w3 = ws[3];
    const float bm = ws[4];
    v2f bmat;
    bmat.x = kh ? w2 : w0;   // VGPR0: K0 (lanes 0-15) | K2 (lanes 16-31)
    bmat.y = kh ? w3 : w1;   // VGPR1: K1 (lanes 0-15) | K3 (lanes 16-31)

    // ---- D = A x B (+0): every column of D equals the 16 dot products ----
    v8f c = {};
    c = __builtin_amdgcn_wmma_f32_16x16x4_f32(
        /*neg_a=*/false, a, /*neg_b=*/false, bmat,
        /*c_mod=*/(short)0, c, /*reuse_a=*/false, /*reuse_b=*/false);

    // ---- Extract column N=0: lane 0 -> M=0..7, lane 16 -> M=8..15 ----
    if (m == 0) {
        float4* op = reinterpret_cast<float4*>(out + base + kh * 8);
        float4 lo, hi;
        lo.x = c[0] + bm; lo.y = c[1] + bm; lo.z = c[2] + bm; lo.w = c[3] + bm;
        hi.x = c[4] + bm; hi.y = c[5] + bm; hi.z = c[6] + bm; hi.w = c[7] + bm;
        op[0] = lo;
        op[1] = hi;
    }
}

// ---------------------------------------------------------------------------
extern "C" void kernel_launch(void* const* d_in, const int* in_sizes, int n_in,
                              void* d_out, int out_size, void* d_ws, size_t ws_size,
                              hipStream_t stream) {
    const float* geom = (const float*)d_in[0];   // (8,256,256,4) fp32
    const float* W    = (const float*)d_in[1];   // (512,4) fp32
    const float* bias = (const float*)d_in[2];   // (512,) fp32
    float*       out  = (float*)d_out;           // (8,256,256) fp32
    float*       ws   = (float*)d_ws;            // >= 5 floats

    // Fold hidden dim: one wave.
    fold_hidden_kernel<<<1, 32, 0, stream>>>(W, bias, ws);

    // Main pass: 524288 outputs, 128 per block -> 4096 blocks of 256 threads.
    const int total  = out_size;                 // 8*256*256 = 524288
    const int blocks = total / 128;              // exactly 4096
    relation_bias_kernel<<<blocks, 256, 0, stream>>>(geom, ws, out);
}